// SelectiveSSMKernel_56530359550298
// MI455X (gfx1250) — compile-verified
//
#include <hip/hip_runtime.h>
#include <hip/hip_bf16.h>

// Selective SSM on gfx1250:
//   1) fused B/C projections via V_WMMA_F32_16X16X4_F32 (u loads shared)
//   2) sequential scan, lane-per-d, state packed in pairs -> v_pk_fma_f32,
//      u tiles staged via GLOBAL_LOAD_ASYNC_TO_LDS (fallback: ds stores)
//   3) in-place RMSNorm over d_model
#define DMODEL 1024
#define NSTATE 64
#define LSEQ   2048
#define NBATCH 4
#define BC_SCALE 0.054126587736527415f        // (1024/3)^-0.5
#define RMS_EPS  1.1920928955078125e-07f

typedef __attribute__((ext_vector_type(2))) float v2f;
typedef __attribute__((ext_vector_type(8))) float v8f;

#if __has_builtin(__builtin_amdgcn_global_load_async_to_lds_b32) && \
    __has_builtin(__builtin_amdgcn_s_wait_asynccnt)
#define HAVE_ASYNC_LDS 1
typedef __attribute__((address_space(1))) int g_i32;   // global (as1)
typedef __attribute__((address_space(3))) int l_i32;   // LDS (as3)
#else
#define HAVE_ASYNC_LDS 0
#endif

// ---------------------------------------------------------------------------
// Kernel 1: Bu[l,n] = scale * sum_d B[n,d]*u[b,d,l]; Cu likewise (fused).
// One wave -> one 16(n) x 16(l) tile for BOTH weights; u loads shared.
// Output layout is [b][l][n] (n contiguous) so the scan's packed operand
// fetches are aligned 8-byte scalar loads.
// ---------------------------------------------------------------------------
__global__ __launch_bounds__(128) void ssm_proj_wmma(
    const float* __restrict__ u,
    const float* __restrict__ Bw,
    const float* __restrict__ Cw,
    float* __restrict__ Bu,
    float* __restrict__ Cu)
{
  const int lane = threadIdx.x & 31;
  const int mt   = threadIdx.x >> 5;   // wave in block -> n-tile 0..3
  const int nt   = blockIdx.x;         // l-tile 0..127
  const int b    = blockIdx.y;         // batch 0..3

  const int M  = lane & 15;
  const int kb = (lane >> 4) << 1;     // K pair for this lane half (0 or 2)

  const float* wpB = Bw + (size_t)(mt * 16 + M) * DMODEL + kb;
  const float* wpC = Cw + (size_t)(mt * 16 + M) * DMODEL + kb;
  const float* upl = u  + (size_t)b * DMODEL * LSEQ
                        + (size_t)kb * LSEQ + (nt * 16 + (lane & 15));

  v8f accB = {}, accC = {};
  #pragma unroll 4
  for (int k = 0; k < DMODEL; k += 4) {
    v2f bb, aB, aC;
    bb[0] = upl[(size_t)k * LSEQ];
    bb[1] = upl[(size_t)(k + 1) * LSEQ];
    aB[0] = wpB[k];  aB[1] = wpB[k + 1];
    aC[0] = wpC[k];  aC[1] = wpC[k + 1];
    accB = __builtin_amdgcn_wmma_f32_16x16x4_f32(false, aB, false, bb,
                                                 (short)0, accB, false, false);
    accC = __builtin_amdgcn_wmma_f32_16x16x4_f32(false, aC, false, bb,
                                                 (short)0, accC, false, false);
  }

  const int ncol  = nt * 16 + (lane & 15);
  const int mbase = mt * 16 + ((lane >> 4) << 3);  // +8 rows for upper lane half
  float* dstB = Bu + ((size_t)b * LSEQ + ncol) * NSTATE + mbase;
  float* dstC = Cu + ((size_t)b * LSEQ + ncol) * NSTATE + mbase;
  #pragma unroll
  for (int v = 0; v < 8; ++v) {        // 8 consecutive floats -> b128 stores
    dstB[v] = accB[v] * BC_SCALE;
    dstC[v] = accC[v] * BC_SCALE;
  }
}

// ---------------------------------------------------------------------------
// Kernel 2: scan. lane = d (32 d / wave); state h packed as 32 x v2f so the
// recurrence runs on v_pk_fma_f32 (2 FLOP/VALU). A/Bu/Cu are wave-uniform ->
// scalar loads (Bu/Cu pairs are contiguous thanks to the [l][n] layout).
// u tile staged through LDS with the async-to-LDS engine when available.
// ---------------------------------------------------------------------------
__global__ __launch_bounds__(128) void ssm_scan(
    const float* __restrict__ u,
    const float* __restrict__ Avec,
    const float* __restrict__ Bu,
    const float* __restrict__ Cu,
    const float* __restrict__ Dvec,
    float* __restrict__ out)
{
  const int wave = threadIdx.x >> 5;
  const int lane = threadIdx.x & 31;
  const int b    = blockIdx.y;
  const int d0   = blockIdx.x * 128 + wave * 32;
  const int dd   = d0 + lane;

  __shared__ float lds_u[4][32][33];   // [wave][d-in-tile][t], padded

  const float* ub  = u + (size_t)b * DMODEL * LSEQ;
  const v2f*   A2  = (const v2f*)Avec;
  const float* buL = Bu + (size_t)b * LSEQ * NSTATE;
  const float* cuL = Cu + (size_t)b * LSEQ * NSTATE;
  float* op = out + ((size_t)b * DMODEL + dd) * LSEQ;
  const float Dd = Dvec[dd];

  v2f h2[NSTATE / 2];
  #pragma unroll
  for (int np = 0; np < NSTATE / 2; ++np) h2[np] = (v2f){0.0f, 0.0f};

  for (int t0 = 0; t0 < LSEQ; t0 += 32) {
    __syncthreads();                              // prev chunk's LDS reads done
#if HAVE_ASYNC_LDS
    #pragma unroll 8
    for (int i = 0; i < 32; ++i) {
      __builtin_amdgcn_global_load_async_to_lds_b32(
          (g_i32*)(ub + (size_t)(d0 + i) * LSEQ + t0 + lane),
          (l_i32*)&lds_u[wave][i][lane],
          0, 0);
    }
    __builtin_amdgcn_s_wait_asynccnt(0);
#else
    #pragma unroll 8
    for (int i = 0; i < 32; ++i)
      lds_u[wave][i][lane] = ub[(size_t)(d0 + i) * LSEQ + t0 + lane];
#endif
    __syncthreads();                              // tile visible (transpose)

    if (t0 + 32 < LSEQ)                           // global_prefetch_b8
      __builtin_prefetch(&ub[(size_t)dd * LSEQ + t0 + 32], 0, 0);

    float uv[32];
    v2f y2[32];
    #pragma unroll
    for (int t = 0; t < 32; ++t) {
      uv[t] = lds_u[wave][lane][t];
      y2[t] = (v2f){0.0f, 0.0f};
    }

    #pragma unroll
    for (int np = 0; np < NSTATE / 2; ++np) {
      const v2f an2 = A2[np];                     // uniform s_load_b64
      const v2f* bup = (const v2f*)(buL + (size_t)t0 * NSTATE) + np;
      const v2f* cup = (const v2f*)(cuL + (size_t)t0 * NSTATE) + np;
      v2f hh = h2[np];
      #pragma unroll
      for (int t = 0; t < 32; ++t) {
        const v2f bu2 = bup[t * (NSTATE / 2)];    // uniform s_load_b64
        const v2f cu2 = cup[t * (NSTATE / 2)];
        const v2f us2 = (v2f){uv[t], uv[t]};
        hh    = hh * an2 + us2 * bu2;             // v_pk_fma_f32 pair
        y2[t] = y2[t] + cu2 * hh;
      }
      h2[np] = hh;
    }

    #pragma unroll
    for (int t = 0; t < 32; ++t)                  // horizontal add + D skip
      op[t0 + t] = (y2[t][0] + y2[t][1]) + uv[t] * Dd;
  }
}

// ---------------------------------------------------------------------------
// Kernel 3: in-place RMSNorm over d_model for each (b, l); thread = l.
// ---------------------------------------------------------------------------
__global__ __launch_bounds__(256) void ssm_rmsnorm(
    float* __restrict__ x, const float* __restrict__ nw)
{
  const int l = blockIdx.x * 256 + threadIdx.x;
  const int b = blockIdx.y;
  float* xp = x + (size_t)b * DMODEL * LSEQ + l;

  float ss = 0.0f;
  #pragma unroll 8
  for (int d = 0; d < DMODEL; ++d) {
    const float v = xp[(size_t)d * LSEQ];
    ss = fmaf(v, v, ss);
  }
  const float r = rsqrtf(ss * (1.0f / (float)DMODEL) + RMS_EPS);
  #pragma unroll 8
  for (int d = 0; d < DMODEL; ++d) {
    xp[(size_t)d * LSEQ] = xp[(size_t)d * LSEQ] * r * nw[d];
  }
}

// ---------------------------------------------------------------------------
// Launcher. Inputs (setup_inputs order): L, u, A, B, C, D, norm_w.
// Workspace: Bu (2 MiB) + Cu (2 MiB) = 4 MiB, layout [b][l][n].
// ---------------------------------------------------------------------------
extern "C" void kernel_launch(void* const* d_in, const int* in_sizes, int n_in,
                              void* d_out, int out_size, void* d_ws, size_t ws_size,
                              hipStream_t stream) {
  (void)in_sizes; (void)n_in; (void)out_size; (void)ws_size;
  const float* u  = (const float*)d_in[1];
  const float* A  = (const float*)d_in[2];
  const float* Bw = (const float*)d_in[3];
  const float* Cw = (const float*)d_in[4];
  const float* Dv = (const float*)d_in[5];
  const float* nw = (const float*)d_in[6];
  float* out = (float*)d_out;

  float* Bu = (float*)d_ws;
  float* Cu = Bu + (size_t)NBATCH * LSEQ * NSTATE;

  dim3 g1(LSEQ / 16, NBATCH);           // l-tiles x batch (B & C fused)
  ssm_proj_wmma<<<g1, 128, 0, stream>>>(u, Bw, Cw, Bu, Cu);

  dim3 g2(DMODEL / 128, NBATCH);        // d-tiles x batch
  ssm_scan<<<g2, 128, 0, stream>>>(u, A, Bu, Cu, Dv, out);

  dim3 g3(LSEQ / 256, NBATCH);          // l-tiles x batch
  ssm_rmsnorm<<<g3, 256, 0, stream>>>(out, nw);
}